// GaussianRegularizationLoss_68573447847949
// MI455X (gfx1250) — compile-verified
//
#include <hip/hip_runtime.h>
#include <math.h>

#define NPTS 8192
#define TILE 16
#define ROWS_PER_WAVE 32
#define WAVES_PER_BLOCK 2
#define ROWS_PER_BLOCK (ROWS_PER_WAVE * WAVES_PER_BLOCK)   // 64
#define NCHUNK 8
#define COLS_PER_CHUNK (NPTS / NCHUNK)                     // 1024
#define TILES_PER_CHUNK (COLS_PER_CHUNK / TILE)            // 64
#define LDS_STRIDE 20   // dwords; 80B row stride -> 16B-aligned b128 reads

typedef float v2f __attribute__((ext_vector_type(2)));
typedef float v8f __attribute__((ext_vector_type(8)));

__device__ __forceinline__ unsigned umin_(unsigned a, unsigned b) { return a < b ? a : b; }
__device__ __forceinline__ unsigned umax_(unsigned a, unsigned b) { return a < b ? b : a; }

// Branchless sorted-insert of x into ascending {k0..k5}, dropping the largest.
// x == 0xFFFFFFFF is a no-op. 11 v_min_u32/v_max_u32, no exec manipulation.
__device__ __forceinline__ void top6_net(unsigned x,
    unsigned &k0, unsigned &k1, unsigned &k2,
    unsigned &k3, unsigned &k4, unsigned &k5)
{
  unsigned t;
  t = umin_(k0, x); x = umax_(k0, x); k0 = t;
  t = umin_(k1, x); x = umax_(k1, x); k1 = t;
  t = umin_(k2, x); x = umax_(k2, x); k2 = t;
  t = umin_(k3, x); x = umax_(k3, x); k3 = t;
  t = umin_(k4, x); x = umax_(k4, x); k4 = t;
  k5 = umin_(k5, x);
}

__device__ __forceinline__ void scan_val(float partial, float sqRow,
    int gcol, int myRow,
    unsigned &k0, unsigned &k1, unsigned &k2,
    unsigned &k3, unsigned &k4, unsigned &k5)
{
  const float d2 = fmaxf(sqRow + partial, 0.0f);
  unsigned key = (__float_as_uint(d2) & 0xFFFFE000u) | (unsigned)gcol;
  key = (gcol == myRow) ? 0xFFFFFFFFu : key;   // self-exclusion, v_cndmask
  top6_net(key, k0, k1, k2, k3, k4, k5);
}

// ---------------------------------------------------------------------------
// Pass 1: WMMA distance tiles + per-lane branchless top-6 over a column chunk.
// grid = (NPTS/ROWS_PER_BLOCK, NCHUNK), block = 64 (2 waves, wave32).
// ---------------------------------------------------------------------------
__global__ __launch_bounds__(ROWS_PER_BLOCK)
void nn_pass1(const float* __restrict__ pos, unsigned* __restrict__ keys_out)
{
  __shared__ float lds[WAVES_PER_BLOCK][ROWS_PER_WAVE * LDS_STRIDE];

  const int lane    = threadIdx.x & 31;
  const int wave    = threadIdx.x >> 5;
  const int halfSel = lane >> 4;      // 0: holds K=0,1 ; 1: holds K=2,3
  const int mIdx    = lane & 15;      // M (for A) / N (for B) within tile
  const int rbase   = blockIdx.x * ROWS_PER_BLOCK + wave * ROWS_PER_WAVE;
  const int myRow   = rbase + lane;   // row this lane owns in the scan phase
  float* const myLds  = lds[wave];
  float* const rowLds = myLds + lane * LDS_STRIDE;

  // A fragments: rows are [x, y, z, 1] (K padded to 4).
  // 32-bit 16x4 A layout: lanes 0-15 = (K0,K1), lanes 16-31 = (K2,K3).
  v2f aTop, aBot;
  {
    const int rT = rbase + mIdx;
    const int rB = rbase + 16 + mIdx;
    const float tx = pos[3*rT + 0], ty = pos[3*rT + 1], tz = pos[3*rT + 2];
    const float bx = pos[3*rB + 0], by = pos[3*rB + 1], bz = pos[3*rB + 2];
    aTop.x = (halfSel == 0) ? tx : tz;
    aTop.y = (halfSel == 0) ? ty : 1.0f;
    aBot.x = (halfSel == 0) ? bx : bz;
    aBot.y = (halfSel == 0) ? by : 1.0f;
  }

  const float px = pos[3*myRow + 0];
  const float py = pos[3*myRow + 1];
  const float pz = pos[3*myRow + 2];
  const float sqRow = px*px + py*py + pz*pz;

  unsigned k0 = 0xFFFFFFFFu, k1 = 0xFFFFFFFFu, k2 = 0xFFFFFFFFu;
  unsigned k3 = 0xFFFFFFFFu, k4 = 0xFFFFFFFFu, k5 = 0xFFFFFFFFu;

  const int cstart = blockIdx.y * COLS_PER_CHUNK;
  for (int t = 0; t < TILES_PER_CHUNK; ++t) {
    const int cbase = cstart + t * TILE;

    // B fragment: column j is [-2x_j, -2y_j, -2z_j, ||p_j||^2] so the WMMA
    // directly produces  ||p_j||^2 - 2 * p_i . p_j  per tile element.
    const int col = cbase + mIdx;
    const float cx = pos[3*col + 0];
    const float cy = pos[3*col + 1];
    const float cz = pos[3*col + 2];
    const float sqCol = cx*cx + cy*cy + cz*cz;
    v2f b;
    b.x = (halfSel == 0) ? (-2.0f * cx) : (-2.0f * cz);   // v_cndmask, no branch
    b.y = (halfSel == 0) ? (-2.0f * cy) : sqCol;

    v8f cT = {}; v8f cB = {};
    cT = __builtin_amdgcn_wmma_f32_16x16x4_f32(false, aTop, false, b,
                                               (short)0, cT, false, false);
    cB = __builtin_amdgcn_wmma_f32_16x16x4_f32(false, aBot, false, b,
                                               (short)0, cB, false, false);

    // C/D layout: VGPR r, lanes 0-15 -> M=r, lanes 16-31 -> M=r+8; N = lane%16.
    #pragma unroll
    for (int r = 0; r < 8; ++r) {
      const int mLoc = r + 8 * halfSel;
      myLds[mLoc        * LDS_STRIDE + mIdx] = cT[r];
      myLds[(16 + mLoc) * LDS_STRIDE + mIdx] = cB[r];
    }
    asm volatile("s_wait_dscnt 0" ::: "memory");  // wave-internal LDS RAW

    // Scan: lane owns full row `lane`; read as 4x b128, process branchlessly.
    #pragma unroll
    for (int q = 0; q < 4; ++q) {
      const float4 v = *(const float4*)(rowLds + 4 * q);
      const int gc = cbase + 4 * q;
      scan_val(v.x, sqRow, gc + 0, myRow, k0, k1, k2, k3, k4, k5);
      scan_val(v.y, sqRow, gc + 1, myRow, k0, k1, k2, k3, k4, k5);
      scan_val(v.z, sqRow, gc + 2, myRow, k0, k1, k2, k3, k4, k5);
      scan_val(v.w, sqRow, gc + 3, myRow, k0, k1, k2, k3, k4, k5);
    }
    asm volatile("s_wait_dscnt 0" ::: "memory");  // keep next stores behind reads
  }

  unsigned* o = keys_out + ((size_t)myRow * NCHUNK + blockIdx.y) * 6;
  o[0] = k0; o[1] = k1; o[2] = k2; o[3] = k3; o[4] = k4; o[5] = k5;
}

// ---------------------------------------------------------------------------
// Pass 2: merge per-chunk top-6 lists; position loss + color smoothness.
// ---------------------------------------------------------------------------
__global__ void nn_merge(const unsigned* __restrict__ keys,
                         const float* __restrict__ pos,
                         const float* __restrict__ colors,
                         float* __restrict__ accum)
{
  const int row = blockIdx.x * blockDim.x + threadIdx.x;
  if (row >= NPTS) return;

  unsigned k0 = 0xFFFFFFFFu, k1 = 0xFFFFFFFFu, k2 = 0xFFFFFFFFu;
  unsigned k3 = 0xFFFFFFFFu, k4 = 0xFFFFFFFFu, k5 = 0xFFFFFFFFu;
  const unsigned* p = keys + (size_t)row * NCHUNK * 6;
  #pragma unroll 6
  for (int i = 0; i < NCHUNK * 6; ++i)
    top6_net(p[i], k0, k1, k2, k3, k4, k5);

  // min_distances = 2nd-smallest off-diagonal distance (k1); recompute exact.
  const float rx = pos[3*row+0], ry = pos[3*row+1], rz = pos[3*row+2];
  {
    const int j = (int)(k1 & 8191u);
    const float dx = rx - pos[3*j+0];
    const float dy = ry - pos[3*j+1];
    const float dz = rz - pos[3*j+2];
    const float mind = sqrtf(fmaxf(dx*dx + dy*dy + dz*dz, 0.0f));
    atomicAdd(&accum[0], expf(-mind));
  }

  // color smoothness over the 5 nearest neighbors (keys k0..k4).
  const float cr = colors[3*row+0], cg = colors[3*row+1], cb = colors[3*row+2];
  const int nn[5] = { (int)(k0 & 8191u), (int)(k1 & 8191u), (int)(k2 & 8191u),
                      (int)(k3 & 8191u), (int)(k4 & 8191u) };
  float s = 0.0f;
  #pragma unroll
  for (int i = 0; i < 5; ++i) {
    const int j = nn[i];
    s += fabsf(cr - colors[3*j+0]) + fabsf(cg - colors[3*j+1]) +
         fabsf(cb - colors[3*j+2]);
  }
  atomicAdd(&accum[1], s);
}

// ---------------------------------------------------------------------------
// Elementwise losses: scales / rotations / colors.
// ---------------------------------------------------------------------------
__global__ void small_losses(const float* __restrict__ scales,
                             const float* __restrict__ rots,
                             const float* __restrict__ colors,
                             float* __restrict__ accum)
{
  const int i = blockIdx.x * blockDim.x + threadIdx.x;
  if (i >= NPTS) return;

  const float s0 = scales[3*i+0], s1 = scales[3*i+1], s2 = scales[3*i+2];
  const float absSum = fabsf(s0 - 1.0f) + fabsf(s1 - 1.0f) + fabsf(s2 - 1.0f);
  const float m = (s0 + s1 + s2) * (1.0f / 3.0f);
  const float v = ((s0-m)*(s0-m) + (s1-m)*(s1-m) + (s2-m)*(s2-m)) * 0.5f; // ddof=1

  const float q0 = rots[4*i+0], q1 = rots[4*i+1], q2 = rots[4*i+2], q3 = rots[4*i+3];
  const float nrm = sqrtf(q0*q0 + q1*q1 + q2*q2 + q3*q3);
  const float rot = (nrm - 1.0f) * (nrm - 1.0f);

  const float c0 = colors[3*i+0] - 0.5f;
  const float c1 = colors[3*i+1] - 0.5f;
  const float c2 = colors[3*i+2] - 0.5f;
  const float col = c0*c0 + c1*c1 + c2*c2;

  atomicAdd(&accum[2], absSum);
  atomicAdd(&accum[3], v);
  atomicAdd(&accum[4], rot);
  atomicAdd(&accum[5], col);
}

__global__ void zero_accum(float* __restrict__ accum)
{
  if (threadIdx.x < 8) accum[threadIdx.x] = 0.0f;
}

__global__ void finalize(const float* __restrict__ accum, float* __restrict__ out)
{
  const float n = (float)NPTS;
  const float position_loss  = accum[0] / n;
  const float scale_total    = accum[2] / (3.0f * n) + accum[3] / n;
  const float rotation_loss  = accum[4] / n;
  const float color_total    = accum[5] / (3.0f * n) + accum[1] / (15.0f * n);
  out[0] = 0.1f * position_loss + 0.1f * scale_total +
           0.1f * rotation_loss + 0.1f * color_total;
}

extern "C" void kernel_launch(void* const* d_in, const int* in_sizes, int n_in,
                              void* d_out, int out_size, void* d_ws, size_t ws_size,
                              hipStream_t stream) {
  const float* positions = (const float*)d_in[0];
  const float* scales    = (const float*)d_in[1];
  const float* rotations = (const float*)d_in[2];
  const float* colors    = (const float*)d_in[3];
  float* out = (float*)d_out;

  float*    accum = (float*)d_ws;                          // 8 floats
  unsigned* keys  = (unsigned*)((char*)d_ws + 256);        // NPTS*NCHUNK*6 u32

  zero_accum<<<1, 32, 0, stream>>>(accum);

  dim3 grid1(NPTS / ROWS_PER_BLOCK, NCHUNK);
  nn_pass1<<<grid1, ROWS_PER_BLOCK, 0, stream>>>(positions, keys);

  small_losses<<<NPTS / 256, 256, 0, stream>>>(scales, rotations, colors, accum);
  nn_merge<<<NPTS / 256, 256, 0, stream>>>(keys, positions, colors, accum);
  finalize<<<1, 1, 0, stream>>>(accum, out);
}